// FlyLoRALinear_2379411882426
// MI455X (gfx1250) — compile-verified
//
#include <hip/hip_runtime.h>

// ---------------------------------------------------------------------------
// FlyLoRA forward for MI455X (gfx1250, wave32):
//   y  = x @ A^T              (M=16384, K=4096, N=32)   -- V_WMMA_F32_16X16X4_F32
//   ay = y * top8(|y+d|) mask (per-token rank over 32)
//   out= ay @ B^T * 2.0       (M=16384, K=32,  N=4096)  -- V_WMMA_F32_16X16X4_F32
// Memory-bound: ~512MB mandatory traffic @ 23.3TB/s => ~22us floor.
// GEMM1 uses GLOBAL_LOAD_ASYNC_TO_LDS_B128 (ASYNCcnt) + double-buffered LDS so
// the async engine streams x while WMMAs consume the previous chunk.
// ---------------------------------------------------------------------------

typedef float v2f __attribute__((ext_vector_type(2)));
typedef float v4f __attribute__((ext_vector_type(4)));
typedef float v8f __attribute__((ext_vector_type(8)));
typedef int   v4i __attribute__((ext_vector_type(4)));

#define IN_F   4096
#define OUT_F  4096
#define RANK   32
#define TOPK   8
#define SCALEF 2.0f
#define LDSPAD 36   // row stride (floats): 16B-aligned rows, conflict-free on 64 banks
#define OTILE  256  // gemm2 outs per workgroup

#if defined(__has_builtin)
#  if __has_builtin(__builtin_amdgcn_global_load_async_to_lds_b128)
#    define HAVE_ASYNC_LDS 1
#  endif
#endif

// Typed, address-space-qualified casts for the async-to-LDS builtin
// (params are v4i pointers in global(1)/shared(3) address spaces).
#define GV4(p) ((__attribute__((address_space(1))) v4i*)(p))
#define LV4(p) ((__attribute__((address_space(3))) v4i*)(p))

__device__ __forceinline__ void wait_async_lds() {
#ifdef HAVE_ASYNC_LDS
#  if __has_builtin(__builtin_amdgcn_s_wait_asynccnt)
    __builtin_amdgcn_s_wait_asynccnt(0);
#  else
    asm volatile("s_wait_asynccnt 0x0" ::: "memory");
#  endif
#endif
}

__device__ __forceinline__ v8f wmma_f32_4(v2f a, v2f b, v8f c) {
    // 8 args: (neg_a, A, neg_b, B, c_mod, C, reuse_a, reuse_b)
    return __builtin_amdgcn_wmma_f32_16x16x4_f32(false, a, false, b, (short)0, c,
                                                 false, false);
}

// ---------------------------------------------------------------------------
// Kernel 1: y[t][r] = sum_k x[t][k] * A[r][k]
// WG = 256 thr (8 waves). Tile: 128 tokens x 32 r, K chunked by 32.
// Double-buffered LDS; chunk k+1 streams in via async-to-LDS while chunk k
// feeds 16 WMMAs per wave. One barrier per chunk.
// ---------------------------------------------------------------------------
__global__ __launch_bounds__(256) void gemm1_xat(const float* __restrict__ x,
                                                 const float* __restrict__ A,
                                                 float* __restrict__ y) {
    __shared__ float sX[2][128][LDSPAD];  // 36KB
    __shared__ float sW[2][32][LDSPAD];   //  9KB

    const int  t     = threadIdx.x;
    const int  lane  = t & 31;
    const int  wid   = t >> 5;            // 0..7
    const int  l15   = lane & 15;
    const int  khalf = (lane >> 4) << 1;  // 0 or 2 (K split across lane halves)
    const long tb    = (long)blockIdx.x * 128;

    const int xr = t >> 3;                // 0..31 row group for cooperative loads
    const int xc = (t & 7) << 2;          // 0,4,...,28

    const float* xrow0 = x + (tb + xr) * (long)IN_F + xc;
    const float* arow  = A + (long)xr * IN_F + xc;

#ifdef HAVE_ASYNC_LDS
#define STAGE1(buf, koff)                                                      \
    do {                                                                       \
        _Pragma("unroll")                                                      \
        for (int i_ = 0; i_ < 4; ++i_)                                         \
            __builtin_amdgcn_global_load_async_to_lds_b128(                    \
                GV4(xrow0 + (long)i_ * 32 * IN_F + (koff)),                    \
                LV4(&sX[buf][xr + i_ * 32][xc]), 0, 0);                        \
        __builtin_amdgcn_global_load_async_to_lds_b128(                        \
            GV4(arow + (koff)), LV4(&sW[buf][xr][xc]), 0, 0);                  \
    } while (0)
#else
#define STAGE1(buf, koff)                                                      \
    do {                                                                       \
        _Pragma("unroll")                                                      \
        for (int i_ = 0; i_ < 4; ++i_)                                         \
            *(v4f*)&sX[buf][xr + i_ * 32][xc] = __builtin_nontemporal_load(    \
                (const v4f*)(xrow0 + (long)i_ * 32 * IN_F + (koff)));          \
        *(v4f*)&sW[buf][xr][xc] = *(const v4f*)(arow + (koff));                \
    } while (0)
#endif

    v8f cLo = {};
    v8f cHi = {};

    STAGE1(0, 0);
    wait_async_lds();
    __syncthreads();

    int cur = 0;
    for (int k0 = 0; k0 < IN_F; k0 += 32) {
        const int nxt = cur ^ 1;
        if (k0 + 32 < IN_F) STAGE1(nxt, k0 + 32);  // prefetch next chunk

        #pragma unroll
        for (int kk = 0; kk < 8; ++kk) {
            const int kb = (kk << 2) + khalf;
            v2f ax  = *(const v2f*)&sX[cur][wid * 16 + l15][kb];  // A-frag: M=token
            v2f wlo = *(const v2f*)&sW[cur][l15][kb];             // B-frag: N=r 0..15
            v2f whi = *(const v2f*)&sW[cur][l15 + 16][kb];        // B-frag: N=r 16..31
            cLo = wmma_f32_4(ax, wlo, cLo);
            cHi = wmma_f32_4(ax, whi, cHi);
        }
        wait_async_lds();
        __syncthreads();
        cur = nxt;
    }

    // C layout: lane L<16 vgpr v = C[M=v][N=L]; lane L+16 vgpr v = C[M=v+8][N=L]
    const long tokBase = tb + wid * 16 + ((lane >> 4) << 3);
    #pragma unroll
    for (int v = 0; v < 8; ++v) {
        y[(tokBase + v) * RANK + l15]      = cLo[v];
        y[(tokBase + v) * RANK + 16 + l15] = cHi[v];
    }
}

// ---------------------------------------------------------------------------
// Kernel 2: per-token top-8 of |y + d| over RANK=32; keep y where selected.
// Rank-by-counting reproduces jax.lax.top_k tie-break (lower index wins).
// ---------------------------------------------------------------------------
__global__ __launch_bounds__(256) void topk_mask(const float* __restrict__ y,
                                                 const float* __restrict__ dbias,
                                                 float* __restrict__ ay, int M) {
    const int tok = blockIdx.x * blockDim.x + threadIdx.x;
    if (tok >= M) return;

    float v[RANK];
    float a[RANK];
    const v4f* yp = (const v4f*)(y + (long)tok * RANK);
    #pragma unroll
    for (int i = 0; i < RANK / 4; ++i) {
        v4f q = yp[i];
        v[i * 4 + 0] = q.x; v[i * 4 + 1] = q.y;
        v[i * 4 + 2] = q.z; v[i * 4 + 3] = q.w;
    }
    #pragma unroll
    for (int i = 0; i < RANK; ++i) a[i] = fabsf(v[i] + dbias[i]);

    #pragma unroll
    for (int i = 0; i < RANK; ++i) {
        int rank = 0;
        for (int j = 0; j < RANK; ++j)
            rank += (a[j] > a[i]) || ((a[j] == a[i]) && (j < i));
        v[i] = (rank < TOPK) ? v[i] : 0.0f;
    }

    v4f* op = (v4f*)(ay + (long)tok * RANK);
    #pragma unroll
    for (int i = 0; i < RANK / 4; ++i) {
        v4f q = { v[i * 4 + 0], v[i * 4 + 1], v[i * 4 + 2], v[i * 4 + 3] };
        op[i] = q;
    }
}

// ---------------------------------------------------------------------------
// Kernel 3: out[t][o] = SCALE * sum_r ay[t][r] * B[o][r]
// WG = 256 thr. Tile: 128 tokens x 256 outs; K=32 fits one LDS stage.
// Wave w: tokens w*16..+15 across 16 N-subtiles (8 hoisted X-frags, 128 WMMAs).
// 256-wide N tile halves the ay re-read (32x -> 16x of 2MB).
// ---------------------------------------------------------------------------
__global__ __launch_bounds__(256) void gemm2_ayb(const float* __restrict__ ay,
                                                 const float* __restrict__ B,
                                                 float* __restrict__ out) {
    __shared__ float sY[128][LDSPAD];     // 18KB
    __shared__ float sB[OTILE][LDSPAD];   // 36KB

    const int  t     = threadIdx.x;
    const int  lane  = t & 31;
    const int  wid   = t >> 5;
    const int  l15   = lane & 15;
    const int  khalf = (lane >> 4) << 1;
    const long tb    = (long)blockIdx.x * 128;
    const long ob    = (long)blockIdx.y * OTILE;

    const int xr = t >> 3;
    const int xc = (t & 7) << 2;

#ifdef HAVE_ASYNC_LDS
    #pragma unroll
    for (int i = 0; i < 4; ++i)
        __builtin_amdgcn_global_load_async_to_lds_b128(
            GV4(ay + (tb + xr + i * 32) * RANK + xc),
            LV4(&sY[xr + i * 32][xc]), 0, 0);
    #pragma unroll
    for (int i = 0; i < 8; ++i)
        __builtin_amdgcn_global_load_async_to_lds_b128(
            GV4(B + (ob + xr + i * 32) * RANK + xc),
            LV4(&sB[xr + i * 32][xc]), 0, 0);
    wait_async_lds();
#else
    #pragma unroll
    for (int i = 0; i < 4; ++i)
        *(v4f*)&sY[xr + i * 32][xc] =
            *(const v4f*)(ay + (tb + xr + i * 32) * RANK + xc);
    #pragma unroll
    for (int i = 0; i < 8; ++i)
        *(v4f*)&sB[xr + i * 32][xc] =
            *(const v4f*)(B + (ob + xr + i * 32) * RANK + xc);
#endif
    __syncthreads();

    // Hoist the 8 K-step X fragments (K=32 => 8 x k4)
    v2f xf[8];
    #pragma unroll
    for (int kk = 0; kk < 8; ++kk)
        xf[kk] = *(const v2f*)&sY[wid * 16 + l15][(kk << 2) + khalf];

    const long tokBase = tb + wid * 16 + ((lane >> 4) << 3);
    #pragma unroll
    for (int nt = 0; nt < OTILE / 16; ++nt) {
        v8f acc = {};
        #pragma unroll
        for (int kk = 0; kk < 8; ++kk) {
            v2f wf = *(const v2f*)&sB[nt * 16 + l15][(kk << 2) + khalf];
            acc = wmma_f32_4(xf[kk], wf, acc);
        }
        const long ocol = ob + nt * 16 + l15;
        #pragma unroll
        for (int v = 0; v < 8; ++v)
            __builtin_nontemporal_store(acc[v] * SCALEF,
                                        out + (tokBase + v) * OUT_F + ocol);
    }
}

// ---------------------------------------------------------------------------
extern "C" void kernel_launch(void* const* d_in, const int* in_sizes, int n_in,
                              void* d_out, int out_size, void* d_ws, size_t ws_size,
                              hipStream_t stream) {
    const float* x  = (const float*)d_in[0];   // [4,4096,4096] f32
    const float* A  = (const float*)d_in[1];   // [32,4096]     f32
    const float* B  = (const float*)d_in[2];   // [4096,32]     f32
    const float* db = (const float*)d_in[3];   // [32]          f32
    float* out = (float*)d_out;                // [4,4096,4096] f32

    const int M = in_sizes[0] / IN_F;          // 16384 tokens

    float* y  = (float*)d_ws;                  // [M,32]  (2MB)
    float* ay = y + (size_t)M * RANK;          // [M,32]  (2MB)

    gemm1_xat<<<M / 128, 256, 0, stream>>>(x, A, y);
    topk_mask<<<(M + 255) / 256, 256, 0, stream>>>(y, db, ay, M);
    gemm2_ayb<<<dim3(M / 128, OUT_F / OTILE), 256, 0, stream>>>(ay, B, out);
}